// LIFNode_82987358093942
// MI455X (gfx1250) — compile-verified
//
#include <hip/hip_runtime.h>

// LIF neuron scan: x[T,B,C,H,W] -> spikes[T,B,C,H,W], T=8, N=B*C*H*W per step.
// Memory-bound streaming kernel (~268 MB total traffic -> ~11.5 us floor at
// 23.3 TB/s). Each thread carries its membrane potential V for 4 sites in
// VGPRs across all 8 timesteps; input is streamed via gfx1250 async
// global->LDS b128 loads (ASYNCcnt split waits, TH=NT streaming hint),
// output via non-temporal b128 stores. Both 134 MB streams bypass temporal
// caching since every byte is touched exactly once.

typedef float v4f __attribute__((ext_vector_type(4)));
// Match the builtin's parameter type: 16-byte int vector pointer.
typedef int   v4i __attribute__((vector_size(16)));

#define T_STEPS 8
#define BLOCK   256
#define VEC     4

// CPol immediate for gfx1250 VMEM: TH field bits [2:0]; TH=1 -> non-temporal.
#define CPOL_NT 1

#if defined(__AMDGCN__) &&                                              \
    __has_builtin(__builtin_amdgcn_global_load_async_to_lds_b128) &&    \
    __has_builtin(__builtin_amdgcn_s_wait_asynccnt)
#define LIF_ASYNC 1
typedef __attribute__((address_space(1))) v4i* glob_v4i_ptr;
typedef __attribute__((address_space(3))) v4i* lds_v4i_ptr;
#else
#define LIF_ASYNC 0
#endif

__global__ __launch_bounds__(BLOCK) void lif_scan_kernel(
    const float* __restrict__ x, float* __restrict__ out, int n_per_t)
{
    const float kInvTau = 1.0f / (5.0f / 3.0f);  // 1/tau_m
    const float kVth    = 1.0f;

    const int tid = (int)threadIdx.x;
    const long long base = ((long long)blockIdx.x * BLOCK + tid) * VEC;
    if (base >= n_per_t) return;   // fully covered: n_per_t % (BLOCK*VEC) == 0

    v4f V = {0.0f, 0.0f, 0.0f, 0.0f};

    // One LIF step on a float4 of input, writes spike vector, updates V.
    auto fire = [&](v4f xv, long long off) {
        v4f ov;
#pragma unroll
        for (int e = 0; e < VEC; ++e) {
            float dv = xv[e] - V[e] * kInvTau;   // -V/tau + x
            float Vn = V[e] + dv;                // integrate
            float o  = (Vn >=  kVth ? 1.0f : 0.0f)
                     - (Vn <= -kVth ? 1.0f : 0.0f);
            ov[e] = o;
            V[e]  = (o == 0.0f) ? Vn : 0.0f;     // hard reset
        }
        __builtin_nontemporal_store(ov, (v4f*)(out + off));  // TH=NT b128 store
    };

#if LIF_ASYNC
    // 8-deep LDS staging (32 KB/WG; 10 WGs still fit per 320 KB WGP).
    // Each lane async-loads its own b128 slot for every timestep up front,
    // then drains in order using split ASYNCcnt waits (async loads complete
    // in issue order). No cross-lane sharing -> no barriers needed.
    __shared__ v4f buf[T_STEPS][BLOCK];

#pragma unroll
    for (int t = 0; t < T_STEPS; ++t) {
        __builtin_amdgcn_global_load_async_to_lds_b128(
            (glob_v4i_ptr)(x + (long long)t * n_per_t + base),
            (lds_v4i_ptr)&buf[t][tid],
            /*imm offset*/ 0, /*cpol*/ CPOL_NT);
    }

#define LIF_STEP(t, remaining)                                          \
    do {                                                                \
        __builtin_amdgcn_s_wait_asynccnt(remaining);                    \
        fire(buf[t][tid], (long long)(t) * n_per_t + base);             \
    } while (0)

    LIF_STEP(0, 7); LIF_STEP(1, 6); LIF_STEP(2, 5); LIF_STEP(3, 4);
    LIF_STEP(4, 3); LIF_STEP(5, 2); LIF_STEP(6, 1); LIF_STEP(7, 0);
#undef LIF_STEP

#else
    // Fallback: direct non-temporal b128 register loads; all 8 issued before
    // the dependent scan so LOADcnt covers the 16MB-strided fetches.
    v4f xs[T_STEPS];
#pragma unroll
    for (int t = 0; t < T_STEPS; ++t)
        xs[t] = __builtin_nontemporal_load(
            (const v4f*)(x + (long long)t * n_per_t + base));
#pragma unroll
    for (int t = 0; t < T_STEPS; ++t)
        fire(xs[t], (long long)t * n_per_t + base);
#endif
}

extern "C" void kernel_launch(void* const* d_in, const int* in_sizes, int n_in,
                              void* d_out, int out_size, void* d_ws, size_t ws_size,
                              hipStream_t stream) {
    const float* x  = (const float*)d_in[0];
    float*      out = (float*)d_out;

    const long long total   = (long long)in_sizes[0];     // T*B*C*H*W
    const int       n_per_t = (int)(total / T_STEPS);     // 4,194,304
    const int       blocks  = (int)((n_per_t + (long long)BLOCK * VEC - 1) /
                                    ((long long)BLOCK * VEC));  // 4096

    lif_scan_kernel<<<blocks, BLOCK, 0, stream>>>(x, out, n_per_t);
}